// MANEAttention_21268678049983
// MI455X (gfx1250) — compile-verified
//
#include <hip/hip_runtime.h>
#include <math.h>

// ---------------------------------------------------------------------------
// MANE multi-network embedding loss for MI455X (gfx1250, wave32).
//
// Loss = weighted sum of log_sigmoid(+/- dot(q,k)) over ~676K gathered
// 128-dim f32 row pairs -> ~350 MB of random 512B-row gathers, ~0.17 GFLOP.
// Memory/latency bound: design maximizes loads-in-flight per wave.
//
// Dots = diag(Q x K^T) via chained V_WMMA_F32_16X16X4_F32 (exact f32).
// A-matrix 16x4 layout (ISA 7.12.2): lane L holds row M=L%16, K-pair
// 2*(L/16) -> one float2 per lane per 4-chunk; B mirrors it.
// Diagonal: lanes 0-7 -> acc[lane], lanes 24-31 -> acc[lane-24].
//
// MLP optimizations: full 16x128 key tile preloaded to regs (32 b64 loads
// in one clause), K-reduction split over 2 independent WMMA accumulators,
// all 11 slot indices hoisted ahead of the slot loop.
// ---------------------------------------------------------------------------

typedef float v2f __attribute__((ext_vector_type(2)));
typedef float v8f __attribute__((ext_vector_type(8)));

#define NUM_NET 3
#define NNODES  100000
#define DD      128
#define BB      4096
#define NSAMP   10

#define W1 (-1.0f / (3.0f * 3.0f * (float)BB))   // cost1 terms
#define W2 (-1.0f / (3.0f * 6.0f * (float)BB))   // cost2 / cost3 off-diagonal terms

#if defined(__gfx1250__) && __has_builtin(__builtin_amdgcn_wmma_f32_16x16x4_f32)
#define HAVE_WMMA_F32 1
#else
#define HAVE_WMMA_F32 0
#endif

__global__ void mane_zero_out(float* out) {
    if (threadIdx.x == 0 && blockIdx.x == 0) out[0] = 0.0f;
}

__device__ __forceinline__ float log_sigmoid_f(float x) {
    // min(x,0) - log1p(exp(-|x|)) : numerically stable for both signs
    return fminf(x, 0.0f) - log1pf(expf(-fabsf(x)));
}

__global__ __launch_bounds__(256) void mane_loss_kernel(
    const float* __restrict__ node_tables,   // (3, N, 128)
    const float* __restrict__ neigh_tables,  // (3, N, 128)
    const int*   __restrict__ nodes_idx,     // (3, B)
    const int*   __restrict__ neigh_idx,     // (3, B)
    const int*   __restrict__ neg1,          // (3, B, NS)
    const int*   __restrict__ neg2,          // (3, 3, B, NS)
    const int*   __restrict__ neg3,          // (3, 3, B, NS)
    float*       __restrict__ out)
{
    const int lane = threadIdx.x & 31;
    const int wave = blockIdx.x * (blockDim.x >> 5) + (threadIdx.x >> 5);

    const int NTILES = BB / 16;                   // 256 b-tiles
    const int TOTAL  = NUM_NET * NTILES * 5;      // 3840 waves
    if (wave >= TOTAL) return;                    // wave-uniform exit

    const int g  = wave % 5;                      // 0: cost1, 1-2: cost2, 3-4: cost3
    const int tb = (wave / 5) % NTILES;
    const int i  = wave / (5 * NTILES);

    const int m = lane & 15;                      // matrix row handled by this lane
    const int h = lane >> 4;                      // K half within a 4-chunk
    const int b = tb * 16 + m;

    // ---- key table / index sources for this group ----
    const float* ktab;
    const int*   pos_idx;
    const int*   negarr;
    float        wgt;
    if (g == 0) {
        ktab    = neigh_tables + (size_t)i * NNODES * DD;
        pos_idx = neigh_idx + i * BB;
        negarr  = neg1 + (size_t)(i * BB) * NSAMP;
        wgt     = W1;
    } else {
        const int j = (i + 1 + ((g - 1) & 1)) % NUM_NET;   // the two j != i
        if (g <= 2) {   // cost2: keys from node_tables[j]
            ktab    = node_tables + (size_t)j * NNODES * DD;
            pos_idx = nodes_idx + i * BB;
            negarr  = neg2 + (size_t)((j * NUM_NET + i) * BB) * NSAMP;
        } else {        // cost3: keys from neigh_tables[j]
            ktab    = neigh_tables + (size_t)j * NNODES * DD;
            pos_idx = neigh_idx + i * BB;
            negarr  = neg3 + (size_t)((j * NUM_NET + i) * BB) * NSAMP;
        }
        wgt = W2;
    }

    // ---- hoist all 11 gather indices (1 + 10 contiguous ints per lane) ----
    int kidxs[11];
    kidxs[0] = pos_idx[b];
    {
        const int* nb = negarr + b * NSAMP;
#pragma unroll
        for (int k = 0; k < NSAMP; ++k) kidxs[1 + k] = nb[k];
    }

    // ---- query: node_v[i, b] preloaded in WMMA A layout (32 float2 chunks) ----
    const int    qidx = nodes_idx[i * BB + b];
    const float* qrow = node_tables + ((size_t)i * NNODES + (size_t)qidx) * DD;
    v2f qa[32];
#pragma unroll
    for (int c = 0; c < 32; ++c)
        qa[c] = *(const v2f*)(qrow + c * 4 + h * 2);

#if HAVE_WMMA_F32
    const bool diagLane = (lane < 8) || (lane >= 24);
    const int  r        = (lane < 8) ? lane : (lane - 24);   // junk for mid lanes (masked)
#else
    const bool diagLane = (lane < 16);
#endif

    float lacc = 0.0f;

    for (int t = 0; t < 11; ++t) {
        const float  sgn  = (t == 0) ? 1.0f : -1.0f;
        const float* krow = ktab + (size_t)kidxs[t] * DD;

        // preload full 16x128 key tile into regs: 32 b64 gathers in flight
        v2f kb[32];
#pragma unroll
        for (int c = 0; c < 32; ++c)
            kb[c] = *(const v2f*)(krow + c * 4 + h * 2);

        float d;
#if HAVE_WMMA_F32
        // two independent accumulator chains (halve WMMA RAW depth)
        v8f a0 = {};
        v8f a1 = {};
#pragma unroll
        for (int c = 0; c < 32; c += 2) {
            a0 = __builtin_amdgcn_wmma_f32_16x16x4_f32(
                false, qa[c],     false, kb[c],     (short)0, a0, false, false);
            a1 = __builtin_amdgcn_wmma_f32_16x16x4_f32(
                false, qa[c + 1], false, kb[c + 1], (short)0, a1, false, false);
        }
        const v8f acc = a0 + a1;   // diag of sum == sum of diags
        // extract diag(Q K^T): lanes 0-7 -> acc[lane]; lanes 24-31 -> acc[lane-24]
        d = acc[0];
        d = (r == 1) ? acc[1] : d;
        d = (r == 2) ? acc[2] : d;
        d = (r == 3) ? acc[3] : d;
        d = (r == 4) ? acc[4] : d;
        d = (r == 5) ? acc[5] : d;
        d = (r == 6) ? acc[6] : d;
        d = (r == 7) ? acc[7] : d;
#else
        // fallback: lane-pair partial dot + xor-16 shuffle combine
        float part = 0.0f;
#pragma unroll
        for (int c = 0; c < 32; ++c) {
            part = fmaf(qa[c].x, kb[c].x, part);
            part = fmaf(qa[c].y, kb[c].y, part);
        }
        d = part + __shfl_xor(part, 16, 32);
#endif
        const float term = wgt * log_sigmoid_f(sgn * d);
        lacc += diagLane ? term : 0.0f;
    }

    // wave32 reduction, then one atomic per wave
#pragma unroll
    for (int off = 16; off > 0; off >>= 1)
        lacc += __shfl_xor(lacc, off, 32);
    if (lane == 0) atomicAdd(out, lacc);
}

extern "C" void kernel_launch(void* const* d_in, const int* in_sizes, int n_in,
                              void* d_out, int out_size, void* d_ws, size_t ws_size,
                              hipStream_t stream) {
    const float* node_tables  = (const float*)d_in[0];
    const float* neigh_tables = (const float*)d_in[1];
    const int*   nodes_idx    = (const int*)d_in[2];
    const int*   neigh_idx    = (const int*)d_in[3];
    const int*   neg1         = (const int*)d_in[4];
    const int*   neg2         = (const int*)d_in[5];
    const int*   neg3         = (const int*)d_in[6];
    float*       out          = (float*)d_out;

    mane_zero_out<<<1, 64, 0, stream>>>(out);

    const int total_waves     = NUM_NET * (BB / 16) * 5;  // 3840
    const int waves_per_block = 8;                        // 256 threads
    const int blocks = (total_waves + waves_per_block - 1) / waves_per_block;
    mane_loss_kernel<<<blocks, 256, 0, stream>>>(
        node_tables, neigh_tables, nodes_idx, neigh_idx, neg1, neg2, neg3, out);
}